// RelGraphConvHetero_38568806318205
// MI455X (gfx1250) — compile-verified
//
#include <hip/hip_runtime.h>

#define N_A   100000
#define N_B   100000
#define EDGES 200000
#define IN_F  256
#define OUT_F 128
#define NREL  3

#define M_BLK 32          // rows per block (100000 % 32 == 0)
#define KC    32          // K-chunk staged in LDS
#define NC    (IN_F / KC) // 8 chunks

typedef __attribute__((ext_vector_type(2))) float v2f;
typedef __attribute__((ext_vector_type(8))) float v8f;
typedef __attribute__((ext_vector_type(4))) int   v4i;

typedef __attribute__((address_space(1))) v4i g_v4i;   // global int4
typedef __attribute__((address_space(3))) v4i l_v4i;   // LDS int4

#if defined(__has_builtin)
#if __has_builtin(__builtin_amdgcn_global_load_async_to_lds_b128)
#define HAVE_ASYNC_LDS 1
#endif
#if __has_builtin(__builtin_amdgcn_s_wait_asynccnt)
#define HAVE_WAIT_ASYNC 1
#endif
#endif

// ---- async copy: 16B global -> LDS (fallback: plain LDS store) ----
__device__ __forceinline__ void async_copy_b128(float* sdst, const float* gsrc) {
#if defined(HAVE_ASYNC_LDS)
  __builtin_amdgcn_global_load_async_to_lds_b128((g_v4i*)gsrc, (l_v4i*)sdst, 0, 0);
#else
  *(float4*)sdst = *(const float4*)gsrc;
#endif
}

__device__ __forceinline__ void wait_async_le5() {
#if defined(HAVE_ASYNC_LDS)
#if defined(HAVE_WAIT_ASYNC)
  __builtin_amdgcn_s_wait_asynccnt(5);
#else
  asm volatile("s_wait_asynccnt 0x5" ::: "memory");
#endif
#endif
}

__device__ __forceinline__ void wait_async_le0() {
#if defined(HAVE_ASYNC_LDS)
#if defined(HAVE_WAIT_ASYNC)
  __builtin_amdgcn_s_wait_asynccnt(0);
#else
  asm volatile("s_wait_asynccnt 0x0" ::: "memory");
#endif
#endif
}

// ---------------- utility kernels ----------------

__global__ void zero_kernel(float* __restrict__ p, long long n) {
  long long i = (long long)blockIdx.x * blockDim.x + threadIdx.x;
  long long stride = (long long)gridDim.x * blockDim.x;
  for (; i < n; i += stride) p[i] = 0.0f;
}

// Wm[r][i][o] = (w_comp[r,0]*weight[0,i,o] + w_comp[r,1]*weight[1,i,o]) * masks[r][o][o]
__global__ void combine_weights_kernel(const float* __restrict__ weight,   // [2,256,128]
                                       const float* __restrict__ w_comp,   // [3,2]
                                       const float* __restrict__ masks,    // [3,128,128]
                                       float* __restrict__ Wm) {           // [3,256,128]
  int idx = blockIdx.x * blockDim.x + threadIdx.x;
  if (idx >= NREL * IN_F * OUT_F) return;
  int r   = idx / (IN_F * OUT_F);
  int rem = idx - r * (IN_F * OUT_F);
  int o   = rem & (OUT_F - 1);
  float w = w_comp[r * 2 + 0] * weight[rem]
          + w_comp[r * 2 + 1] * weight[IN_F * OUT_F + rem];
  float d = masks[r * OUT_F * OUT_F + o * OUT_F + o];
  Wm[idx] = w * d;
}

__global__ void count_kernel(const int* __restrict__ dst, float* __restrict__ cnt, int n_edges) {
  int e = blockIdx.x * blockDim.x + threadIdx.x;
  if (e < n_edges) atomicAdd(&cnt[dst[e]], 1.0f);
}

__global__ void rcnt_kernel(float* __restrict__ cnt, int n) {
  int i = blockIdx.x * blockDim.x + threadIdx.x;
  if (i < n) cnt[i] = 1.0f / fmaxf(cnt[i], 1.0f);
}

// ---------------- fp32 WMMA GEMM with async-LDS staged panels ----------------
// H[rows,128] = X[rows,256] @ W[256,128]; FINISH adds aggregated out + bias, ReLU in place.
// Block: 256 threads = 8 waves; tile 32 rows x 128 cols; wave w -> col tile w, 2 row tiles.
// K staged in 8 chunks of 32 via GLOBAL_LOAD_ASYNC_TO_LDS_B128, double buffered.
// A 16x4 frag: lanes 0-15 -> (M=lane, K=k+{0,1}); lanes 16-31 -> (M=lane-16, K=k+{2,3})
// B 4x16 frag: lanes 0-15 -> (K=k+{0,1}, N=lane); lanes 16-31 -> (K=k+{2,3}, N=lane-16)
// C/D 16x16:   VGPR v: lanes 0-15 -> M=v, lanes 16-31 -> M=v+8; N = lane&15
template <bool FINISH>
__global__ void wmma_gemm_t(const float* __restrict__ X,
                            const float* __restrict__ W,
                            const float* __restrict__ bias,
                            float* __restrict__ H) {
  __shared__ float sA[2][M_BLK * KC];    // 2 x 4 KB
  __shared__ float sW[2][KC * OUT_F];    // 2 x 16 KB

  const int tid  = threadIdx.x;
  const int lane = tid & 31;
  const int wave = tid >> 5;
  const int row0 = blockIdx.x * M_BLK;
  const int col  = (wave << 4) + (lane & 15);
  const int half = lane >> 4;        // 0: lanes 0-15, 1: lanes 16-31
  const int l15  = lane & 15;

  const float* Xblk = X + (long long)row0 * IN_F;

  // ---- stage chunk 0 ----
  {
#pragma unroll
    for (int i = 0; i < 4; ++i) {
      int j = tid + i * 256;                       // W chunk: 1024 float4
      async_copy_b128(&sW[0][j * 4], W + j * 4);
    }
    int r = tid >> 3, s = tid & 7;                 // A chunk: 256 float4
    async_copy_b128(&sA[0][r * KC + s * 4], Xblk + (long long)r * IN_F + s * 4);
  }

  v8f acc[2];
  acc[0] = {};
  acc[1] = {};

  for (int c = 0; c < NC; ++c) {
    const int cur = c & 1;
    if (c + 1 < NC) {
      const int nxt = cur ^ 1;
      const int kc  = (c + 1) * KC;
#pragma unroll
      for (int i = 0; i < 4; ++i) {
        int j = tid + i * 256;
        async_copy_b128(&sW[nxt][j * 4], W + kc * OUT_F + j * 4);
      }
      int r = tid >> 3, s = tid & 7;
      async_copy_b128(&sA[nxt][r * KC + s * 4], Xblk + (long long)r * IN_F + kc + s * 4);
      wait_async_le5();   // previous chunk's 5 async ops retired
    } else {
      wait_async_le0();
    }
    __syncthreads();

    const float* sWb = &sW[cur][0];
    const float* sAb = &sA[cur][0];
#pragma unroll
    for (int ks = 0; ks < KC; ks += 4) {
      v2f b;
      b.x = sWb[(ks + (half << 1)) * OUT_F + col];
      b.y = sWb[(ks + 1 + (half << 1)) * OUT_F + col];
#pragma unroll
      for (int t = 0; t < 2; ++t) {
        v2f a = *(const v2f*)&sAb[(t * 16 + l15) * KC + ks + (half << 1)];
        acc[t] = __builtin_amdgcn_wmma_f32_16x16x4_f32(false, a, false, b, (short)0,
                                                       acc[t], false, false);
      }
    }
    __syncthreads();   // protect buffer reuse
  }

  // ---- epilogue ----
  float bcol = 0.0f;
  if (FINISH) bcol = bias[col];
#pragma unroll
  for (int t = 0; t < 2; ++t) {
    float* hp = H + (long long)(row0 + t * 16 + (half << 3)) * OUT_F + col;
    if (FINISH) {
#pragma unroll
      for (int v = 0; v < 8; ++v) {
        float val = acc[t][v] + hp[(long long)v * OUT_F] + bcol;
        hp[(long long)v * OUT_F] = fmaxf(val, 0.0f);
      }
    } else {
#pragma unroll
      for (int v = 0; v < 8; ++v) hp[(long long)v * OUT_F] = acc[t][v];
    }
  }
}

// ---------------- edge scatter: out[dst] += h[src] * rcnt[dst] (float4 per thread) ----------------
__global__ void scatter_kernel(const float* __restrict__ Hsrc,
                               const int* __restrict__ src,
                               const int* __restrict__ dst,
                               const float* __restrict__ rcnt,
                               float* __restrict__ out,
                               int n_edges) {
  long long tid   = (long long)blockIdx.x * blockDim.x + threadIdx.x;
  long long total = (long long)n_edges * (OUT_F / 4);
  if (tid >= total) return;
  int e = (int)(tid >> 5);             // 32 threads per edge, 4 floats each
  int o = ((int)tid & 31) << 2;
  int s = src[e], d = dst[e];
  float r = rcnt[d];
  const float4 hv = *(const float4*)(Hsrc + (long long)s * OUT_F + o);
  float* op = out + (long long)d * OUT_F + o;
  atomicAdd(op + 0, hv.x * r);
  atomicAdd(op + 1, hv.y * r);
  atomicAdd(op + 2, hv.z * r);
  atomicAdd(op + 3, hv.w * r);
}

// ---------------- launch ----------------

extern "C" void kernel_launch(void* const* d_in, const int* in_sizes, int n_in,
                              void* d_out, int out_size, void* d_ws, size_t ws_size,
                              hipStream_t stream) {
  const float* x_a    = (const float*)d_in[0];
  const float* x_b    = (const float*)d_in[1];
  const float* weight = (const float*)d_in[2];
  const float* w_comp = (const float*)d_in[3];
  const float* masks  = (const float*)d_in[4];
  const float* h_bias = (const float*)d_in[5];
  const float* loop_w = (const float*)d_in[6];
  const int*   src0   = (const int*)d_in[7];
  const int*   dst0   = (const int*)d_in[8];
  const int*   src1   = (const int*)d_in[9];
  const int*   dst1   = (const int*)d_in[10];
  const int*   src2   = (const int*)d_in[11];
  const int*   dst2   = (const int*)d_in[12];

  float* out   = (float*)d_out;
  float* out_a = out;                                   // [N_A,128]
  float* out_b = out + (long long)N_A * OUT_F;          // [N_B,128]

  // workspace (floats): Wm[3*256*128] | h[100000*128] | cnt0[N_B] cnt1[N_A] cnt2[N_A]
  float* ws   = (float*)d_ws;
  float* Wm   = ws;
  float* h    = Wm + NREL * IN_F * OUT_F;
  float* cnt0 = h + (long long)N_A * OUT_F;
  float* cnt1 = cnt0 + N_B;
  float* cnt2 = cnt1 + N_A;

  const int T = 256;

  zero_kernel<<<2048, T, 0, stream>>>(out, (long long)(N_A + N_B) * OUT_F);
  zero_kernel<<<512, T, 0, stream>>>(cnt0, (long long)(N_B + 2 * N_A));

  combine_weights_kernel<<<(NREL * IN_F * OUT_F + T - 1) / T, T, 0, stream>>>(
      weight, w_comp, masks, Wm);

  const int eb = (EDGES + T - 1) / T;
  count_kernel<<<eb, T, 0, stream>>>(dst0, cnt0, EDGES);
  count_kernel<<<eb, T, 0, stream>>>(dst1, cnt1, EDGES);
  count_kernel<<<eb, T, 0, stream>>>(dst2, cnt2, EDGES);
  rcnt_kernel<<<((N_B + 2 * N_A) + T - 1) / T, T, 0, stream>>>(cnt0, N_B + 2 * N_A);

  const int sb = (int)(((long long)EDGES * (OUT_F / 4) + T - 1) / T);

  // rel0: A -> B
  wmma_gemm_t<false><<<N_A / M_BLK, T, 0, stream>>>(x_a, Wm + 0 * IN_F * OUT_F, nullptr, h);
  scatter_kernel<<<sb, T, 0, stream>>>(h, src0, dst0, cnt0, out_b, EDGES);
  // rel1: B -> A
  wmma_gemm_t<false><<<N_B / M_BLK, T, 0, stream>>>(x_b, Wm + 1 * IN_F * OUT_F, nullptr, h);
  scatter_kernel<<<sb, T, 0, stream>>>(h, src1, dst1, cnt1, out_a, EDGES);
  // rel2: A -> A
  wmma_gemm_t<false><<<N_A / M_BLK, T, 0, stream>>>(x_a, Wm + 2 * IN_F * OUT_F, nullptr, h);
  scatter_kernel<<<sb, T, 0, stream>>>(h, src2, dst2, cnt2, out_a, EDGES);

  // self-loop GEMM + bias + ReLU epilogue (reads aggregated out tile as addend)
  wmma_gemm_t<true><<<N_A / M_BLK, T, 0, stream>>>(x_a, loop_w, h_bias, out_a);
  wmma_gemm_t<true><<<N_B / M_BLK, T, 0, stream>>>(x_b, loop_w, h_bias, out_b);
}